// LinearPredictiveCoefficientsToLineSpectralPairs_88708254532039
// MI455X (gfx1250) — compile-verified
//
#include <hip/hip_runtime.h>
#include <hip/hip_bf16.h>
#include <math.h>

typedef __attribute__((ext_vector_type(2))) float v2f;
typedef __attribute__((ext_vector_type(8))) float v8f;

#define NF    16      // frames per block (one wave32)
#define LGRID 1025    // grid points
#define NPAIR 1024    // adjacent pairs
#define MORD  64      // LPC order == roots kept per frame
#define KPAD  36      // 33 Chebyshev coefs padded to 9 wmma k-steps of 4

__global__ __launch_bounds__(32)
void lpc2lsp_wmma_kernel(const float* __restrict__ a,
                         const float* __restrict__ Tx,
                         const float* __restrict__ x,
                         float* __restrict__ out,
                         int nframes) {
  __shared__ float abuf[NF * 65];      // staged LPC vectors
  __shared__ float c1s[NF][KPAD];      // symmetric-poly Chebyshev coefs (reversed, padded)
  __shared__ float c2s[NF][KPAD];      // antisymmetric-poly coefs
  __shared__ float Kbuf[NF];
  __shared__ float ybt1[NF][20];       // y1 window: cols [c0-4 .. c0+15]
  __shared__ float ybt2[NF][20];       // y2 window
  __shared__ int   posL[NF][MORD];     // per-frame ordered root-cell list
  __shared__ float yupL[NF][MORD];
  __shared__ float ylowL[NF][MORD];
  __shared__ int   cntL[NF];
  __shared__ float padUp[NF], padLow[NF];   // values at pair 1023 for padding path

  const int lane   = threadIdx.x;           // wave32: 0..31
  const int frame0 = blockIdx.x * NF;

  // ---------------- Phase 0a: coalesced stage of 16 frames of `a` ----------
  for (int i = lane; i < NF * 65; i += 32) {
    const long gi = (long)frame0 * 65 + i;
    abuf[i] = (gi < (long)nframes * 65) ? a[gi] : 0.0f;
  }
  __syncthreads();

  // ---------------- Phase 0b: coefficient transform (lane f -> frame f) ----
  // d1=[1,q1], d2=[1,q2]; sign[i]=(i even?-1:+1); cumsum(d*sign[1:]) == -cumsum(d*sign[:-1])
  // => c1[i] = sign[i]*cumsum(d1*sign)[i]; c2[i] = cumsum(d2)[i]; then reverse.
  if (lane < NF) {
    const int f = lane;
    float cs_a = 0.0f, cs2 = 0.0f;
#pragma unroll
    for (int i = 0; i <= 32; ++i) {
      float d1i, d2i;
      if (i == 0) { d1i = 1.0f; d2i = 1.0f; }
      else {
        const float p1v = abuf[f * 65 + i];        // ar[i-1]
        const float p2v = abuf[f * 65 + 65 - i];   // ar[64-i] (reversed half)
        d1i = p1v + p2v; d2i = p1v - p2v;
      }
      const float sgn = (i & 1) ? 1.0f : -1.0f;
      cs_a += d1i * sgn;
      cs2  += d2i;
      c1s[f][32 - i] = sgn * cs_a;                 // store reversed
      c2s[f][32 - i] = cs2;
    }
#pragma unroll
    for (int i = 33; i < KPAD; ++i) { c1s[f][i] = 0.0f; c2s[f][i] = 0.0f; }
    Kbuf[f] = abuf[f * 65];
    cntL[f] = 0;
  }
  __syncthreads();

  // ---------------- A-fragments hoisted to registers -----------------------
  // 16x4 f32 A layout: lanes 0-15 hold M=lane, K={4k,4k+1}; lanes 16-31 K={4k+2,4k+3}
  const int fA   = lane & 15;
  const int kOff = (lane >> 4) * 2;
  float a1f[18], a2f[18];
#pragma unroll
  for (int ks = 0; ks < 9; ++ks) {
    a1f[2 * ks]     = c1s[fA][4 * ks + kOff];
    a1f[2 * ks + 1] = c1s[fA][4 * ks + kOff + 1];
    a2f[2 * ks]     = c2s[fA][4 * ks + kOff];
    a2f[2 * ks + 1] = c2s[fA][4 * ks + kOff + 1];
  }

  // Branchless B addressing: clamp rows >=34 to 33 (their A coefs are zero, so
  // the loaded value is multiplied by 0 inside the WMMA; only the address must
  // be valid). Offsets are per-lane constants across all tiles.
  int offx[9], offy[9];
#pragma unroll
  for (int ks = 0; ks < 9; ++ks) {
    const int r0 = 4 * ks + kOff;
    const int rx = (r0     < 34) ? r0     : 33;
    const int ry = (r0 + 1 < 34) ? r0 + 1 : 33;
    offx[ks] = rx * LGRID;
    offy[ks] = ry * LGRID;
  }

  bool prev1 = false, prev2 = false;  // index1/2 at pair p-1 (scan state, lanes<16)
  int  cnt   = 0;

  // ---------------- Main loop: 65 tiles of 16 grid columns -----------------
  for (int t = 0; t < 65; ++t) {
    const int c0  = 16 * t;
    const int col = c0 + (lane & 15);
    // clamp col for the final partial tile; garbage y-cols >=1025 are never
    // consumed (scan skips p>1023 and gates the p=1023 lookahead flags)
    const int ccl = (col < LGRID) ? col : (LGRID - 1);

    // prefetch next tile's Tx columns (global_prefetch_b8)
    if (t < 64) {
      const int pc = (col + 16 < LGRID) ? col + 16 : LGRID - 1;
      __builtin_prefetch(&Tx[kOff * LGRID + pc], 0, 1);
    }

    // gather all 18 B values first (clusters the global loads)
    v2f bfr[9];
#pragma unroll
    for (int ks = 0; ks < 9; ++ks) {
      bfr[ks].x = Tx[offx[ks] + ccl];
      bfr[ks].y = Tx[offy[ks] + ccl];
    }

    v8f acc1 = {0.f,0.f,0.f,0.f,0.f,0.f,0.f,0.f};
    v8f acc2 = {0.f,0.f,0.f,0.f,0.f,0.f,0.f,0.f};
#pragma unroll
    for (int ks = 0; ks < 9; ++ks) {
      v2f a1; a1.x = a1f[2 * ks]; a1.y = a1f[2 * ks + 1];
      v2f a2; a2.x = a2f[2 * ks]; a2.y = a2f[2 * ks + 1];
      acc1 = __builtin_amdgcn_wmma_f32_16x16x4_f32(false, a1, false, bfr[ks],
                                                   (short)0, acc1, false, false);
      acc2 = __builtin_amdgcn_wmma_f32_16x16x4_f32(false, a2, false, bfr[ks],
                                                   (short)0, acc2, false, false);
    }

    // scatter C tile into LDS window cols 4..19 (C: N=lane%16, M=r + 8*(lane/16))
    const int ci = 4 + (lane & 15);
    const int fb = (lane >> 4) * 8;
#pragma unroll
    for (int r = 0; r < 8; ++r) {
      ybt1[fb + r][ci] = acc1[r];
      ybt2[fb + r][ci] = acc2[r];
    }
    __syncthreads();

    // ordered sign-change scan: finalize pairs [c0-2 .. c0+13] (lane f -> frame f)
    if (lane < NF) {
      const int f = lane;
#pragma unroll 1
      for (int k = 0; k < 16; ++k) {
        const int p = c0 - 2 + k;
        if (p < 0 || p > NPAIR - 1) continue;
        const int bi = p - c0 + 4;                 // 2..17, window index of col p
        const float y10 = ybt1[f][bi], y11 = ybt1[f][bi + 1], y12 = ybt1[f][bi + 2];
        const float y20 = ybt2[f][bi], y21 = ybt2[f][bi + 1], y22 = ybt2[f][bi + 2];
        const bool i1c = (y10 * y11) <= 0.0f;                       // index1[p]
        const bool i2c = (y20 * y21) <= 0.0f;                       // index2[p]
        const bool i1n = (p < NPAIR - 1) && ((y11 * y12) <= 0.0f);  // index1[p+1]
        const bool i2n = (p < NPAIR - 1) && ((y21 * y22) <= 0.0f);
        // widened masks: i[j] = index[j] | index[j-1] (edges handled by false)
        const float yup  = ((i1c || prev1) ? y10 : 0.0f) + ((i2c || prev2) ? y20 : 0.0f);
        const float ylow = ((i1n || i1c)  ? y11 : 0.0f) + ((i2n || i2c)  ? y21 : 0.0f);
        if ((i1c || i2c) && cnt < MORD) {
          posL[f][cnt] = p; yupL[f][cnt] = yup; ylowL[f][cnt] = ylow; ++cnt;
        }
        if (p == NPAIR - 1) { padUp[f] = yup; padLow[f] = ylow; }   // padding path
        prev1 = i1c; prev2 = i2c;
      }
      // carry last 4 columns to window front for next tile
#pragma unroll
      for (int j = 0; j < 4; ++j) {
        ybt1[f][j] = ybt1[f][16 + j];
        ybt2[f][j] = ybt2[f][16 + j];
      }
    }
    __syncthreads();
  }

  if (lane < NF) cntL[lane] = cnt;
  __syncthreads();

  // ---------------- Final phase: interpolate roots + arccos, write out -----
  for (int o = lane; o < NF * 65; o += 32) {
    const int f = o / 65;
    const int s = o - f * 65;
    const int gframe = frame0 + f;
    if (gframe >= nframes) continue;
    float val;
    if (s == 0) {
      val = Kbuf[f];
    } else {
      const int r = s - 1;
      int p; float yup, ylow;
      if (r < cntL[f]) { p = posL[f][r]; yup = yupL[f][r]; ylow = ylowL[f][r]; }
      else             { p = NPAIR - 1;  yup = padUp[f];   ylow = padLow[f];   }
      const float xu = x[p], xl = x[p + 1];
      float xr = (ylow * xu - yup * xl) / (ylow - yup);
      xr = fminf(1.0f, fmaxf(-1.0f, xr));
      val = acosf(xr);
    }
    out[(size_t)gframe * 65 + s] = val;
  }
}

extern "C" void kernel_launch(void* const* d_in, const int* in_sizes, int n_in,
                              void* d_out, int out_size, void* d_ws, size_t ws_size,
                              hipStream_t stream) {
  const float* a  = (const float*)d_in[0];   // (B,T,65) f32
  const float* Tx = (const float*)d_in[1];   // (34,1025) f32
  const float* x  = (const float*)d_in[2];   // (1025,) f32
  float* out = (float*)d_out;                // (B,T,65) f32
  const int nframes = in_sizes[0] / 65;      // 32768
  const int blocks  = (nframes + NF - 1) / NF;
  lpc2lsp_wmma_kernel<<<blocks, 32, 0, stream>>>(a, Tx, x, out, nframes);
}